// blendv2c_39453569581122
// MI455X (gfx1250) — compile-verified
//
#include <hip/hip_runtime.h>
#include <math.h>
#include <stdint.h>

// ---- problem constants (from reference) ----
constexpr int   kT  = 730;
constexpr int   kG  = 2000;
constexpr int   kM  = 4;
constexpr int   kP  = 31;
constexpr int   kHP = kP * kM + 4;     // 128 floats per g in hybrid_params
constexpr float kNZ = 1e-5f;
constexpr int   kUH = 15;

// ---- scan kernel tiling ----
constexpr int kGB        = 64;          // g cells per block
constexpr int kThreads   = 256;         // = kGB * kM  (lane = (g_local, m))
constexpr int kF4        = kGB * 9;     // 576 float4 of weights per step per block (36 f32/g)
constexpr int kF4PerWave = kF4 / 8;     // 72
constexpr int kIssues    = 3;           // ceil(72/32) async b128 issues per wave per step

__constant__ float c_lb[kP] = {0.0f,0.5f,0.001f,0.3f,0.05f,20.0f,0.001f,1.0f,0.1f,1.0f,
                               0.1f,1.0f,5.0f,0.1f,1.0f,0.0f,1.0f,0.1f,0.1f,20.0f,
                               0.0001f,0.0001f,1.0f,0.1f,1.0f,0.0f,-2.0f,-5.0f,0.0f,0.05f,10.0f};
__constant__ float c_ub[kP] = {1.0f,3.0f,3.0f,1.0f,0.5f,300.0f,0.5f,5.0f,50.0f,5.0f,
                               50.0f,5.0f,10.0f,50.0f,5.0f,1.0f,50.0f,50.0f,50.0f,300.0f,
                               0.1f,0.1f,5.0f,50.0f,5.0f,20.0f,3.0f,2.0f,1.0f,0.4f,500.0f};

__device__ __forceinline__ float sigm(float x) { return 1.0f / (1.0f + expf(-x)); }

__device__ __forceinline__ float getp(const float* __restrict__ hp, int g, int m, int idx) {
  float raw = hp[(size_t)g * kHP + idx * kM + m];
  return c_lb[idx] + (c_ub[idx] - c_lb[idx]) * sigm(raw);
}

__device__ __forceinline__ float sum4(float v) {   // sum over the 4 m-lanes (wave32)
  v += __shfl_xor(v, 1, 32);
  v += __shfl_xor(v, 2, 32);
  return v;
}

__device__ __forceinline__ void softmax3(const float* __restrict__ wr, int m, int j, float* o) {
  float a0 = wr[(j * 3 + 0) * 4 + m];
  float a1 = wr[(j * 3 + 1) * 4 + m];
  float a2 = wr[(j * 3 + 2) * 4 + m];
  float mx = fmaxf(a0, fmaxf(a1, a2));
  float e0 = expf(a0 - mx), e1 = expf(a1 - mx), e2 = expf(a2 - mx);
  float inv = 1.0f / (e0 + e1 + e2);
  o[0] = e0 * inv; o[1] = e1 * inv; o[2] = e2 * inv;
}

// =====================================================================
// Kernel 1: sequential scan over T, one lane per (g, m).
// Weights streamed global->LDS via CDNA5 async DMA, double buffered.
// =====================================================================
__global__ __launch_bounds__(kThreads)
void hydro_scan(const float* __restrict__ xphy,     // (T,G,3)
                const float* __restrict__ wts,      // (T,G,36) raw weight logits
                const float* __restrict__ hp,       // (G,128)
                float* __restrict__ out,            // Q | aet | vad | phr, each (T,G)
                float* __restrict__ qk_ws,
                float* __restrict__ qb_ws)
{
  __shared__ __align__(16) float lds_w[2][kGB * 36];

  const int tid  = threadIdx.x;
  const int lane = tid & 31;
  const int wv   = tid >> 5;
  const int m    = tid & 3;
  const int gl   = tid >> 2;
  const int gb   = blockIdx.x * kGB;
  const int ng   = min(kGB, kG - gb);
  const int nf4  = ng * 9;
  int g = gb + gl;
  const bool active = (g < kG);
  if (!active) g = kG - 1;

  // Issue the per-step weight block (ng*144 bytes, contiguous) as async b128
  // copies into LDS. Exactly kIssues per wave (index-clamped duplicates are
  // byte-identical writes), so s_wait_asynccnt immediates are uniform.
  auto issue = [&](int t) {
    const uint64_t src = (uint64_t)(uintptr_t)wts + ((uint64_t)t * kG + (uint64_t)gb) * 144ull;
    float* db = lds_w[t & 1];
#pragma unroll
    for (int it = 0; it < kIssues; ++it) {
      int u = wv * kF4PerWave + it * 32 + lane;
      u = (u < nf4) ? u : (nf4 - 1);
      uint64_t ga = src + (uint64_t)u * 16ull;
      uint32_t la = (uint32_t)(uintptr_t)(db + (size_t)u * 4);
      asm volatile("global_load_async_to_lds_b128 %0, %1, off"
                   :: "v"(la), "v"(ga) : "memory");
    }
  };

  // -------- bounded physical parameters (kept in VGPRs) --------
  const float p_beta   = getp(hp, g, m, 1);    // inf_hbv_beta
  const float p_bexp   = getp(hp, g, m, 2);    // inf_vic_bexp
  const float p_alpha  = getp(hp, g, m, 3);    // inf_hmets_alpha
  const float p_x3a    = getp(hp, g, m, 5);    // bf1_gr4j_x3
  const float p_plbfc  = getp(hp, g, m, 6);    // bf1_pl_bfc
  const float p_plbfn  = getp(hp, g, m, 7);    // bf1_pl_bfn
  const float p_vbmax1 = getp(hp, g, m, 8);    // bf1_vic_bfmax
  const float p_vbn1   = getp(hp, g, m, 9);    // bf1_vic_bfn
  const float p_mperc  = getp(hp, g, m, 16);   // perc_gawser_maxperc
  const float p_crise  = getp(hp, g, m, 17);   // crise_hbv_crise
  const float p_cbmax  = getp(hp, g, m, 18);   // bf2_con_bfmax
  const float p_x3b    = getp(hp, g, m, 19);   // bf2_gr4j_x3
  const float p_lbfc   = getp(hp, g, m, 20);   // bf2_lin_bfc
  const float p_vbmax2 = getp(hp, g, m, 23);   // bf2_vic_bfmax
  const float p_vbn2   = getp(hp, g, m, 24);   // bf2_vic_bfn
  const float p_ddf    = getp(hp, g, m, 25);
  const float p_tbm    = getp(hp, g, m, 26);
  const float p_tbf    = getp(hp, g, m, 27);
  const float p_cfr    = getp(hp, g, m, 28);
  const float p_swi    = getp(hp, g, m, 29);
  const float p_vmax   = getp(hp, g, m, 30);   // vadose_max_level

  // -------- state --------
  float sn0 = kNZ, sn1 = kNZ, sn2 = kNZ;
  float lq0 = kNZ, lq1 = kNZ, lq2 = kNZ;
  float cm0 = kNZ, cm1 = kNZ, cm2 = kNZ;
  float vad = 0.5f * p_vmax, phr = kNZ;

  issue(0);
  float nxp = xphy[(size_t)g * 3 + 0];
  float nxt = xphy[(size_t)g * 3 + 1];
  float nxe = xphy[(size_t)g * 3 + 2];

  for (int t = 0; t < kT; ++t) {
    if (t + 1 < kT) {
      issue(t + 1);                                   // prefetch next step
      asm volatile("s_wait_asynccnt 0x3" ::: "memory"); // this step's copies done
    } else {
      asm volatile("s_wait_asynccnt 0x0" ::: "memory");
    }
    __syncthreads();                                  // block-wide visibility

    const float prcp = nxp, temp = nxt, pet = nxe;    // register-pipelined forcing
    if (t + 1 < kT) {
      const size_t xb = ((size_t)(t + 1) * kG + g) * 3;
      nxp = xphy[xb + 0]; nxt = xphy[xb + 1]; nxe = xphy[xb + 2];
    }

    const float* wr = &lds_w[t & 1][gl * 36];
    float w0[3], w1[3], w2[3];
    softmax3(wr, m, 0, w0);
    softmax3(wr, m, 1, w1);
    softmax3(wr, m, 2, w2);

    // ---- snow module (3 bands) ----
    const float rain = (temp >= 0.0f) ? prcp : 0.0f;
    const float snf  = (temp <  0.0f) ? prcp : 0.0f;
    sn0 += snf; sn1 += snf; sn2 += snf;
    const float fac2 = fminf(1.0f + cm2 * 0.01f, 2.0f);
    const float mb  = p_ddf * fmaxf(temp - p_tbm, 0.0f);
    const float ml0 = fminf(mb, sn0), ml1 = fminf(mb, sn1), ml2 = fminf(mb * fac2, sn2);
    const float rb  = p_cfr * p_ddf * fmaxf(p_tbf - temp, 0.0f);
    const float rf0 = fminf(rb, lq0), rf1 = fminf(rb, lq1), rf2 = fminf(rb, lq2);
    sn0 += rf0 - ml0; sn1 += rf1 - ml1; sn2 += rf2 - ml2;
    lq0 += ml0 - rf0; lq1 += ml1 - rf1; lq2 += ml2 - rf2;
    const float cp0 = p_swi * sn0, cp1 = p_swi * sn1, cp2 = p_swi * sn2;
    const float re0 = fmaxf(lq0 - cp0, 0.0f);
    const float re1 = fmaxf(lq1 - cp1, 0.0f);
    const float re2 = fmaxf(lq2 - cp2, 0.0f);
    lq0 = fminf(lq0, cp0); lq1 = fminf(lq1, cp1); lq2 = fminf(lq2, cp2);
    cm0 = (sn0 > 10.0f * kNZ) ? cm0 + ml0 : 0.0f;
    cm1 = (sn1 > 10.0f * kNZ) ? cm1 + ml1 : 0.0f;
    cm2 = (sn2 > 10.0f * kNZ) ? cm2 + ml2 : 0.0f;
    const float water = rain + (re0 + re1 + re2) * (1.0f / 3.0f);

    // ---- vadose zone ----
    const float sm = fminf(fmaxf(vad / p_vmax, 1e-4f), 1.0f - 1e-4f);
    const float i1 = p_alpha * water * (1.0f - sm);
    const float i2 = water * (1.0f - powf(sm, p_beta));
    const float i3 = water * powf(1.0f - sm, p_bexp);
    const float infil = w0[0] * i1 + w0[1] * i2 + w0[2] * i3;
    float runoff = fmaxf(water - infil, 0.0f);
    const float rr = vad / p_x3a;
    const float r4 = (rr * rr) * (rr * rr);
    const float q1 = vad * (1.0f - powf(1.0f + r4, -0.25f));
    const float q2 = p_plbfc * p_vmax * powf(sm, p_plbfn);
    const float q3 = p_vbmax1 * powf(sm, p_vbn1);
    const float bf1 = fminf(w1[0] * q1 + w1[1] * q2 + w1[2] * q3, vad);
    const float perc  = p_mperc * sm;
    const float crise = fminf(p_crise * (1.0f - sm), phr);
    const float aet   = pet * sm;
    const float vn    = vad + infil + crise - bf1 - perc - aet;
    const float over  = fmaxf(vn - p_vmax, 0.0f);
    vad = fminf(fmaxf(vn, kNZ), p_vmax);
    runoff += over;

    // ---- phreatic zone ----
    const float c1 = fminf(p_cbmax, phr);
    const float c2 = p_lbfc * phr;
    const float c3 = p_vbmax2 * powf(phr / (phr + p_x3b), p_vbn2);
    const float bf2 = fminf(w2[0] * c1 + w2[1] * c2 + w2[2] * c3, phr);
    phr = fmaxf(phr + perc - crise - bf2, kNZ);

    // ---- outputs: means over M via wave32 shuffles ----
    const float s_qk = sum4(runoff)    * 0.25f;
    const float s_qb = sum4(bf1 + bf2) * 0.25f;
    const float s_ae = sum4(aet)       * 0.25f;
    const float s_vd = sum4(vad)       * 0.25f;
    const float s_ph = sum4(phr)       * 0.25f;
    if (m == 0 && active) {
      const size_t o = (size_t)t * kG + g;
      qk_ws[o] = s_qk;
      qb_ws[o] = s_qb;
      out[(size_t)1 * kT * kG + o] = s_ae;
      out[(size_t)2 * kT * kG + o] = s_vd;
      out[(size_t)3 * kT * kG + o] = s_ph;
    }
    __syncthreads();   // protect buf[t&1] before it is overwritten at t+2
  }
}

// =====================================================================
// Kernel 2: gamma unit-hydrograph routing (15-tap causal depthwise FIR)
// =====================================================================
__device__ __forceinline__ void uh_gamma_dev(float a, float th_, float* w) {
  const float aa = fmaxf(a, 0.0f) + 0.1f;
  const float th = fmaxf(th_, 0.0f) + 0.5f;
  const float lg = lgammaf(aa);
  const float lth = logf(th);
  float s = 0.0f;
#pragma unroll
  for (int i = 0; i < kUH; ++i) {
    const float tt = (float)i + 0.5f;
    const float lw = (aa - 1.0f) * logf(tt) - tt / th - lg - aa * lth;
    w[i] = expf(lw);
    s += w[i];
  }
  const float inv = 1.0f / s;
#pragma unroll
  for (int i = 0; i < kUH; ++i) w[i] *= inv;
}

__global__ __launch_bounds__(256)
void hydro_route(const float* __restrict__ hp,
                 const float* __restrict__ qk,
                 const float* __restrict__ qb,
                 float* __restrict__ Qout)
{
  const int g = blockIdx.x * 256 + threadIdx.x;
  if (g >= kG) return;
  const int t0 = blockIdx.y * 73;                 // 730 = 10 * 73
  const float rlb[4] = {0.3f, 0.01f, 0.5f, 0.15f};
  const float rub[4] = {20.0f, 5.0f, 13.0f, 1.5f};
  float rp[4];
#pragma unroll
  for (int r = 0; r < 4; ++r)
    rp[r] = rlb[r] + (rub[r] - rlb[r]) * sigm(hp[(size_t)g * kHP + kP * kM + r]);

  float uh1[kUH], uh2[kUH];
  uh_gamma_dev(rp[0], rp[1], uh1);
  uh_gamma_dev(rp[2], rp[3], uh2);

  const int t1 = min(t0 + 73, kT);
  for (int t = t0; t < t1; ++t) {
    float acc = 0.0f;
#pragma unroll
    for (int l = 0; l < kUH; ++l) {
      const int tt = t - l;
      if (tt >= 0)
        acc += uh1[l] * qk[(size_t)tt * kG + g] + uh2[l] * qb[(size_t)tt * kG + g];
    }
    Qout[(size_t)t * kG + g] = acc;
  }
}

// =====================================================================
extern "C" void kernel_launch(void* const* d_in, const int* in_sizes, int n_in,
                              void* d_out, int out_size, void* d_ws, size_t ws_size,
                              hipStream_t stream) {
  (void)in_sizes; (void)n_in; (void)out_size; (void)ws_size;
  const float* xphy = (const float*)d_in[0];   // (T,G,3)
  const float* wts  = (const float*)d_in[1];   // (T,G,36)
  const float* hp   = (const float*)d_in[2];   // (G,128)
  float* out = (float*)d_out;                  // [Q | aet | vad | phr], each T*G
  float* qk  = (float*)d_ws;                   // scratch: (T,G)
  float* qb  = qk + (size_t)kT * kG;           // scratch: (T,G)

  const int nblk = (kG + kGB - 1) / kGB;       // 32
  hydro_scan<<<nblk, kThreads, 0, stream>>>(xphy, wts, hp, out, qk, qb);

  dim3 rg((kG + 255) / 256, 10);
  hydro_route<<<rg, 256, 0, stream>>>(hp, qk, qb, out);
}